// HQNNQuanv_64312840290383
// MI455X (gfx1250) — compile-verified
//
#include <hip/hip_runtime.h>
#include <hip/hip_bf16.h>

// Sizes fixed by the reference: B=1024, H=28, K=2 -> F=27, S=F*F=729, NQ=4,
// n_feat = 4*729 = 2916, output (1024,10).
#define NB     1024
#define FDIM   27
#define SPP    (FDIM * FDIM)       // 729 patches per image
#define NFEAT  (4 * SPP)           // 2916
#define NOUT   10

typedef __attribute__((ext_vector_type(2))) float v2f;
typedef __attribute__((ext_vector_type(8))) float v8f;

// ---------------------------------------------------------------------------
// Kernel 1: per-patch 4-qubit statevector sim -> <Z_q> features, written
// K-major: feat_t[(q*729 + s) * 1024 + b]  (coalesced across b).
// Supports arbitrary circuit weights (reference happens to pass zeros).
// State index: idx = b0*8 + b1*4 + b2*2 + b3  (qubit0 = bit3 ... qubit3 = bit0)
// ---------------------------------------------------------------------------
__global__ void quanv_features_kernel(const float* __restrict__ x,
                                      const float* __restrict__ wts,
                                      float* __restrict__ feat_t) {
    const int t = blockIdx.x * blockDim.x + threadIdx.x;
    if (t >= SPP * NB) return;
    const int b = t & (NB - 1);       // batch index (coalesced)
    const int s = t >> 10;            // spatial patch index 0..728
    const int fi = s / FDIM;
    const int fj = s - fi * FDIM;

    const float* __restrict__ img = x + (size_t)b * 28 * 28;
    const float p0 = img[(fi    ) * 28 + fj    ];
    const float p1 = img[(fi    ) * 28 + fj + 1];
    const float p2 = img[(fi + 1) * 28 + fj    ];
    const float p3 = img[(fi + 1) * 28 + fj + 1];

    const float HPI = 1.57079632679489662f;   // pi/2 (angle = pi*p, use p*pi/2)
    float cq[4], sq[4];
    cq[0] = __cosf(HPI * p0); sq[0] = __sinf(HPI * p0);
    cq[1] = __cosf(HPI * p1); sq[1] = __sinf(HPI * p1);
    cq[2] = __cosf(HPI * p2); sq[2] = __sinf(HPI * p2);
    cq[3] = __cosf(HPI * p3); sq[3] = __sinf(HPI * p3);

    float re[16], im[16];
#pragma unroll
    for (int i = 0; i < 16; ++i) {
        float a = ((i >> 3) & 1) ? sq[0] : cq[0];
        a      *= ((i >> 2) & 1) ? sq[1] : cq[1];
        a      *= ((i >> 1) & 1) ? sq[2] : cq[2];
        a      *= ( i       & 1) ? sq[3] : cq[3];
        re[i] = a;
        im[i] = 0.0f;
    }

    const float w0 = wts[0], w1 = wts[1], w2 = wts[2], w3 = wts[3];
    const float cx0 = __cosf(0.5f * w0), sx0 = __sinf(0.5f * w0);
    const float cx1 = __cosf(0.5f * w1), sx1 = __sinf(0.5f * w1);
    const float cy2 = __cosf(0.5f * w2), sy2 = __sinf(0.5f * w2);
    const float cy3 = __cosf(0.5f * w3), sy3 = __sinf(0.5f * w3);

    // RX pair update: new_a = cw*a - i*sw*b ; new_b = -i*sw*a + cw*b
#define RX_PAIR(A, B, CW, SW)                                   \
    {                                                           \
        float ra = re[A], ia = im[A], rb = re[B], ib = im[B];   \
        re[A] =  CW * ra + SW * ib;                             \
        im[A] =  CW * ia - SW * rb;                             \
        re[B] =  SW * ia + CW * rb;                             \
        im[B] = -SW * ra + CW * ib;                             \
    }
    // RY pair update: new_a = cw*a - sw*b ; new_b = sw*a + cw*b (real matrix)
#define RY_PAIR(A, B, CW, SW)                                   \
    {                                                           \
        float ra = re[A], ia = im[A], rb = re[B], ib = im[B];   \
        re[A] = CW * ra - SW * rb;                              \
        im[A] = CW * ia - SW * ib;                              \
        re[B] = SW * ra + CW * rb;                              \
        im[B] = SW * ia + CW * ib;                              \
    }
#define SWAP_AMP(A, B)                                          \
    {                                                           \
        float tr = re[A]; re[A] = re[B]; re[B] = tr;            \
        float ti = im[A]; im[A] = im[B]; im[B] = ti;            \
    }

    // RX(w0) on qubit0 (bit 3)
#pragma unroll
    for (int i = 0; i < 8; ++i) RX_PAIR(i, i + 8, cx0, sx0);
    // RX(w1) on qubit1 (bit 2)
#pragma unroll
    for (int i = 0; i < 16; ++i)
        if (!(i & 4)) RX_PAIR(i, i + 4, cx1, sx1);
    // CNOT(2,3): control bit1, target bit0
#pragma unroll
    for (int i = 0; i < 16; ++i)
        if ((i & 2) && !(i & 1)) SWAP_AMP(i, i + 1);
    // CNOT(0,2): control bit3, target bit1
#pragma unroll
    for (int i = 0; i < 16; ++i)
        if ((i & 8) && !(i & 2)) SWAP_AMP(i, i + 2);
    // CNOT(0,3): control bit3, target bit0
#pragma unroll
    for (int i = 0; i < 16; ++i)
        if ((i & 8) && !(i & 1)) SWAP_AMP(i, i + 1);
    // RY(w2) on qubit0 (bit 3)
#pragma unroll
    for (int i = 0; i < 8; ++i) RY_PAIR(i, i + 8, cy2, sy2);
    // RY(w3) on qubit3 (bit 0)
#pragma unroll
    for (int i = 0; i < 16; i += 2) RY_PAIR(i, i + 1, cy3, sy3);

    float ev0 = 0.f, ev1 = 0.f, ev2 = 0.f, ev3 = 0.f;
#pragma unroll
    for (int i = 0; i < 16; ++i) {
        const float pr = re[i] * re[i] + im[i] * im[i];
        ev0 += (i & 8) ? -pr : pr;
        ev1 += (i & 4) ? -pr : pr;
        ev2 += (i & 2) ? -pr : pr;
        ev3 += (i & 1) ? -pr : pr;
    }

    feat_t[(size_t)(0 * SPP + s) * NB + b] = ev0;
    feat_t[(size_t)(1 * SPP + s) * NB + b] = ev1;
    feat_t[(size_t)(2 * SPP + s) * NB + b] = ev2;
    feat_t[(size_t)(3 * SPP + s) * NB + b] = ev3;
#undef RX_PAIR
#undef RY_PAIR
#undef SWAP_AMP
}

// ---------------------------------------------------------------------------
// Kernel 2: pack W (10 x 2916, row-major) into Wt (2916 x 16, zero-padded N)
// so WMMA B-fragment loads are contiguous.
// ---------------------------------------------------------------------------
__global__ void pack_w_kernel(const float* __restrict__ W,
                              float* __restrict__ Wt) {
    const int t = blockIdx.x * blockDim.x + threadIdx.x;
    if (t >= NFEAT * 16) return;
    const int k = t >> 4;
    const int n = t & 15;
    Wt[t] = (n < NOUT) ? W[(size_t)n * NFEAT + k] : 0.0f;
}

// ---------------------------------------------------------------------------
// Kernel 3: out(1024,10) = feat(1024,2916) @ W^T + b via V_WMMA_F32_16X16X4_F32.
// One wave per 16-row M-tile (64 blocks x 32 threads). N=16 single tile.
// A-frag (16x4 f32): lanes 0-15 -> K=0/1, lanes 16-31 -> K=2/3, M = lane&15.
// D (16x16 f32): VGPR v, lanes 0-15 -> M=v, lanes 16-31 -> M=8+v, N = lane&15.
// ---------------------------------------------------------------------------
__global__ void wmma_gemm_kernel(const float* __restrict__ feat_t,
                                 const float* __restrict__ Wt,
                                 const float* __restrict__ bias,
                                 float* __restrict__ out) {
    const int lane = threadIdx.x;         // 0..31
    const int half = lane >> 4;           // 0: K=0/1, 1: K=2/3
    const int l    = lane & 15;
    const int m0   = blockIdx.x * 16;     // M-tile base row

    v8f acc = {};
    for (int k = 0; k < NFEAT; k += 4) {
        const int ka = k + half * 2;
        v2f a, bb;
        a.x  = feat_t[(size_t)(ka    ) * NB + m0 + l];
        a.y  = feat_t[(size_t)(ka + 1) * NB + m0 + l];
        bb.x = Wt[(ka    ) * 16 + l];
        bb.y = Wt[(ka + 1) * 16 + l];
#if __has_builtin(__builtin_amdgcn_wmma_f32_16x16x4_f32)
        acc = __builtin_amdgcn_wmma_f32_16x16x4_f32(
            /*neg_a=*/false, a, /*neg_b=*/false, bb,
            /*c_mod=*/(short)0, acc, /*reuse_a=*/false, /*reuse_b=*/false);
#else
        // Fallback (should not be taken on gfx1250): scalar rank-4 update of
        // this lane's slice of D. Kept only so the TU always compiles.
        #pragma unroll
        for (int v = 0; v < 8; ++v) acc[v] += a.x * bb.x + a.y * bb.y;
#endif
    }

    const int n    = l;                   // output column
    const int mrow = m0 + half * 8;       // lanes 16-31 hold M = 8..15 of tile
    if (n < NOUT) {
#pragma unroll
        for (int v = 0; v < 8; ++v) {
            out[(size_t)(mrow + v) * NOUT + n] = acc[v] + bias[n];
        }
    }
}

// ---------------------------------------------------------------------------
// Launch: inputs in setup_inputs() order: x, W, b, weights (all float32).
// Workspace: feat_t (2916*1024 f32) | Wt (2916*16 f32)  ~= 12.1 MB.
// ---------------------------------------------------------------------------
extern "C" void kernel_launch(void* const* d_in, const int* in_sizes, int n_in,
                              void* d_out, int out_size, void* d_ws, size_t ws_size,
                              hipStream_t stream) {
    const float* x    = (const float*)d_in[0];   // (1024,1,28,28)
    const float* W    = (const float*)d_in[1];   // (10,2916)
    const float* bias = (const float*)d_in[2];   // (10,)
    const float* wts  = (const float*)d_in[3];   // (1,4) circuit weights
    float* out = (float*)d_out;                  // (1024,10)

    float* feat_t = (float*)d_ws;                          // 2916*1024 floats
    float* Wt     = feat_t + (size_t)NFEAT * NB;           // 2916*16 floats

    {   // features
        const int total = SPP * NB;                        // 746496
        const int block = 256;
        quanv_features_kernel<<<(total + block - 1) / block, block, 0, stream>>>(
            x, wts, feat_t);
    }
    {   // pack W
        const int total = NFEAT * 16;                      // 46656
        const int block = 256;
        pack_w_kernel<<<(total + block - 1) / block, block, 0, stream>>>(W, Wt);
    }
    {   // WMMA GEMM + bias
        wmma_gemm_kernel<<<NB / 16, 32, 0, stream>>>(feat_t, Wt, bias, out);
    }
}